// CIN_52853867544735
// MI455X (gfx1250) — compile-verified
//
#include <hip/hip_runtime.h>

// CIN (xDeepFM-style) with per-slot attention on CDNA5 f32 WMMA.
// GEMM per layer:  [B*K=8192, 40*Fi] @ [40*Fi, 200]  (f32, matches reference)
// v4: packed row-pair B matrix (1 b64 load per B-frag), fused {q,max,1/sum}
//     float4 stat stream (1 b128 load per fill element), LDS ping-pong
//     (1 barrier per chunk, fill overlaps WMMA across waves).

typedef float v2f __attribute__((ext_vector_type(2)));
typedef float v4f __attribute__((ext_vector_type(4)));
typedef float v8f __attribute__((ext_vector_type(8)));

#define B_    256
#define F0_   40
#define K_    32
#define FNEXT 200
#define NR    8192          // B_*K_ output rows r = b*32+k

// dst[b,f,k] = src[b,f,k] * w[f,k]   (WLEN = F*K, compile-time)
template <int WLEN>
__global__ void ew_mul_kernel(float* __restrict__ dst, const float* __restrict__ src,
                              const float* __restrict__ w, int n) {
  int i = blockIdx.x * blockDim.x + threadIdx.x;
  if (i < n) dst[i] = src[i] * w[i % WLEN];
}

// Pack cw row-pairs: P[c/2][j] = {cw[c][j], cw[c+1][j]}  -> b64 B-frag loads.
template <int KDIM>
__global__ void pack_cw_kernel(v2f* __restrict__ P, const float* __restrict__ cw) {
  int idx = blockIdx.x * blockDim.x + threadIdx.x;   // (KDIM/2)*FNEXT threads
  int cp = idx / FNEXT;
  int j  = idx - cp * FNEXT;
  v2f p;
  p.x = cw[(2 * cp) * FNEXT + j];
  p.y = cw[(2 * cp + 1) * FNEXT + j];
  P[idx] = p;
}

// Fused per-(g,r) stream: T[g*8192 + r] = {q, softmax-max, 1/softmax-sum, 0}.
// q = res[b,g,k] * wq[g,k];  max/sum over f of A[b,f,k]*q.
template <int FI>
__global__ void softmax_denom_kernel(v4f* __restrict__ T,
                                     const float* __restrict__ A,
                                     const float* __restrict__ res,
                                     const float* __restrict__ wq) {
  int idx = blockIdx.x * blockDim.x + threadIdx.x;   // FI*8192 threads
  int g = idx >> 13;
  int r = idx & (NR - 1);
  int b = r >> 5, k = r & 31;
  float q = res[(b * FI + g) * K_ + k] * wq[g * K_ + k];
  float av[F0_];
#pragma unroll
  for (int f = 0; f < F0_; ++f) av[f] = A[(b * F0_ + f) * K_ + k];
  float m = -3.4e38f;
#pragma unroll
  for (int f = 0; f < F0_; ++f) m = fmaxf(m, av[f] * q);
  float s = 0.f;
#pragma unroll
  for (int f = 0; f < F0_; ++f) s += __expf(av[f] * q - m);
  v4f t; t.x = q; t.y = m; t.z = 1.0f / s; t.w = 0.f;
  T[idx] = t;
}

// Fused X-generation + f32 WMMA GEMM, 64 output rows per block.
// 256 threads = 8 waves; wave w owns column tiles w and w+8 (13 cover N=200).
template <int FI, int NSPLIT>
__global__ __launch_bounds__(256)
void cin_gemm_kernel(float* __restrict__ Rbase,
                     const float* __restrict__ A, const float* __restrict__ V,
                     const v4f* __restrict__ T,
                     const v2f* __restrict__ P) {
  constexpr int KDIM   = F0_ * FI;           // 1600 or 8000
  constexpr int KSLICE = KDIM / NSPLIT;      // 1600 or 4000
  constexpr int KC     = 160;                // divides both KSLICEs
  constexpr int SEG    = KC / 4;             // 40 elems per fill thread
  constexpr int LDSS   = KC + 4;             // 164: conflict-free b64 frag reads
  constexpr int NCH    = KSLICE / KC;
  __shared__ float Xs[2][64 * LDSS];         // ping-pong, 2 x 41 KB

  const int tid   = threadIdx.x;
  const int wave  = tid >> 5;
  const int lane  = tid & 31;
  const int rblk  = blockIdx.x & 127;
  const int ks    = blockIdx.x >> 7;
  const int rbase = rblk * 64;
  const int kbase = ks * KSLICE;
  float* __restrict__ R = Rbase + ks * (B_ * FNEXT * K_);

  const int mrow = lane & 15;
  const int koff = (lane < 16) ? 0 : 2;
  const int j0   = wave * 16 + mrow;         // tile t0 = wave (< 200 always)
  const int j1   = (wave + 8) * 16 + mrow;   // tile t1 = wave + 8
  const bool t1v = (wave + 8) < 13;

  // fill role: 4 threads per row, SEG consecutive c each
  const int fm = tid & 63;
  const int fr = rbase + fm;
  const int fb = fr >> 5, fk = fr & 31;
  const int ccbeg = (tid >> 6) * SEG;

  v8f acc0[4] = {};
  v8f acc1[4] = {};

  auto fill = [&](int ck0, float* xb) {
    int cpos = ck0 + ccbeg;
    int f = cpos / FI;
    int g = cpos - f * FI;
    float a = A[(fb * F0_ + f) * K_ + fk];
    float v = V[(fb * F0_ + f) * K_ + fk];
    const v4f* tp = T + (g << 13) + fr;
    float* xp = xb + fm * LDSS + ccbeg;
    for (int i = 0; i < SEG; i += 4) {
      v4f xv;
#pragma unroll
      for (int u = 0; u < 4; ++u) {
        v4f t = tp[0];
        xv[u] = __expf(a * t.x - t.y) * t.z * v;
        tp += NR;
        if (++g == FI) {                      // wave-uniform wrap
          g = 0; ++f;
          tp -= FI * NR;
          a = A[(fb * F0_ + f) * K_ + fk];
          v = V[(fb * F0_ + f) * K_ + fk];
        }
      }
      *(v4f*)(xp + i) = xv;
    }
  };

  fill(kbase, Xs[0]);
  __syncthreads();

  for (int ci = 0; ci < NCH; ++ci) {
    const int ck0 = kbase + ci * KC;
    const float* xb = Xs[ci & 1];
    // ---- WMMA: 4 A-frags (b64 LDS) + 1-2 B-frags (b64 global) per step ----
    if (t1v) {
      const v2f* bq0 = P + ((ck0 + koff) >> 1) * FNEXT + j0;
      const v2f* bq1 = P + ((ck0 + koff) >> 1) * FNEXT + j1;
      for (int cc = 0; cc < KC; cc += 4) {
        v2f a[4];
#pragma unroll
        for (int rt = 0; rt < 4; ++rt) {
          const float* xp = &xb[(rt * 16 + mrow) * LDSS + cc + koff];
          a[rt].x = xp[0]; a[rt].y = xp[1];
        }
        v2f b0 = bq0[0];
        v2f b1 = bq1[0];
#pragma unroll
        for (int rt = 0; rt < 4; ++rt)
          acc0[rt] = __builtin_amdgcn_wmma_f32_16x16x4_f32(
                         false, a[rt], false, b0, (short)0, acc0[rt], false, false);
#pragma unroll
        for (int rt = 0; rt < 4; ++rt)
          acc1[rt] = __builtin_amdgcn_wmma_f32_16x16x4_f32(
                         false, a[rt], false, b1, (short)0, acc1[rt], false, false);
        bq0 += 2 * FNEXT; bq1 += 2 * FNEXT;
      }
    } else {
      const v2f* bq0 = P + ((ck0 + koff) >> 1) * FNEXT + j0;
      for (int cc = 0; cc < KC; cc += 4) {
        v2f a[4];
#pragma unroll
        for (int rt = 0; rt < 4; ++rt) {
          const float* xp = &xb[(rt * 16 + mrow) * LDSS + cc + koff];
          a[rt].x = xp[0]; a[rt].y = xp[1];
        }
        v2f b0 = bq0[0];
#pragma unroll
        for (int rt = 0; rt < 4; ++rt)
          acc0[rt] = __builtin_amdgcn_wmma_f32_16x16x4_f32(
                         false, a[rt], false, b0, (short)0, acc0[rt], false, false);
        bq0 += 2 * FNEXT;
      }
    }
    // fill next chunk into the other buffer (overlaps WMMA across waves)
    if (ci + 1 < NCH) fill(kbase + (ci + 1) * KC, Xs[(ci + 1) & 1]);
    __syncthreads();
  }

  // ---- epilogue: rows of a tile map to 8 consecutive k -> b128 stores ----
  const int madd = (lane < 16) ? 0 : 8;
#pragma unroll
  for (int rt = 0; rt < 4; ++rt) {
    int r0 = rbase + rt * 16 + madd;         // v=0 row; k consecutive over v
    int b = r0 >> 5, k0 = r0 & 31;
    float* d0 = &R[(b * FNEXT + j0) * K_ + k0];
    v4f lo = {acc0[rt][0], acc0[rt][1], acc0[rt][2], acc0[rt][3]};
    v4f hi = {acc0[rt][4], acc0[rt][5], acc0[rt][6], acc0[rt][7]};
    *(v4f*)d0 = lo; *(v4f*)(d0 + 4) = hi;
    if (t1v && j1 < FNEXT) {
      float* d1 = &R[(b * FNEXT + j1) * K_ + k0];
      v4f lo1 = {acc1[rt][0], acc1[rt][1], acc1[rt][2], acc1[rt][3]};
      v4f hi1 = {acc1[rt][4], acc1[rt][5], acc1[rt][6], acc1[rt][7]};
      *(v4f*)d1 = lo1; *(v4f*)(d1 + 4) = hi1;
    }
  }
}

// d_out[b, j] = sum_k of layer outputs; layer-1 comes in 2 split-K partials.
__global__ void reduce_k_kernel(float* __restrict__ out, const float* __restrict__ R1,
                                const float* __restrict__ R2a, const float* __restrict__ R2b) {
  int idx = blockIdx.x * blockDim.x + threadIdx.x;   // B*400 threads
  int j = idx % (2 * FNEXT);
  int b = idx / (2 * FNEXT);
  float s = 0.f;
  if (j < FNEXT) {
    const v4f* p = (const v4f*)(R1 + (b * FNEXT + j) * K_);
#pragma unroll
    for (int k = 0; k < K_ / 4; ++k) { v4f t = p[k]; s += t.x + t.y + t.z + t.w; }
  } else {
    const v4f* pa = (const v4f*)(R2a + (b * FNEXT + (j - FNEXT)) * K_);
    const v4f* pb = (const v4f*)(R2b + (b * FNEXT + (j - FNEXT)) * K_);
#pragma unroll
    for (int k = 0; k < K_ / 4; ++k) {
      v4f ta = pa[k], tb = pb[k];
      s += ta.x + ta.y + ta.z + ta.w + tb.x + tb.y + tb.z + tb.w;
    }
  }
  out[idx] = s;
}

extern "C" void kernel_launch(void* const* d_in, const int* in_sizes, int n_in,
                              void* d_out, int out_size, void* d_ws, size_t ws_size,
                              hipStream_t stream) {
  const float* inputs = (const float*)d_in[0];
  const float* wq0 = (const float*)d_in[1];
  const float* wk0 = (const float*)d_in[2];
  const float* wv0 = (const float*)d_in[3];
  const float* cw0 = (const float*)d_in[4];
  const float* wq1 = (const float*)d_in[5];
  const float* wk1 = (const float*)d_in[6];
  const float* wv1 = (const float*)d_in[7];
  const float* cw1 = (const float*)d_in[8];

  float* ws = (float*)d_ws;
  const int n0 = B_ * F0_ * K_;            // 327680
  const int nQ = B_ * FNEXT * K_;          // 1638400
  float* A  = ws;        ws += n0;
  float* V  = ws;        ws += n0;
  v4f*   T  = (v4f*)ws;  ws += 4 * nQ;     // {q, max, 1/sum, 0} per (g, r)
  float* R1 = ws;        ws += nQ;
  float* R2 = ws;        ws += 2 * nQ;     // split-K partials
  v2f*   P  = (v2f*)ws;  ws += 2 * ((F0_ * FNEXT / 2) * FNEXT); // ~55 MB total

  dim3 blk(256);
  // ---- layer 0 (Fi = 40, no split-K) ----
  ew_mul_kernel<F0_ * K_><<<n0 / 256, blk, 0, stream>>>(A, inputs, wk0, n0);
  ew_mul_kernel<F0_ * K_><<<n0 / 256, blk, 0, stream>>>(V, inputs, wv0, n0);
  softmax_denom_kernel<F0_><<<(F0_ * NR) / 256, blk, 0, stream>>>(T, A, inputs, wq0);
  pack_cw_kernel<F0_ * F0_><<<(F0_ * F0_ / 2) * FNEXT / 256, blk, 0, stream>>>(P, cw0);
  cin_gemm_kernel<F0_, 1><<<NR / 64, blk, 0, stream>>>(R1, A, V, T, P);
  // ---- layer 1 (Fi = 200, split-K = 2) ----
  ew_mul_kernel<F0_ * K_><<<n0 / 256, blk, 0, stream>>>(A, inputs, wk1, n0);
  ew_mul_kernel<F0_ * K_><<<n0 / 256, blk, 0, stream>>>(V, inputs, wv1, n0);
  softmax_denom_kernel<FNEXT><<<(FNEXT * NR) / 256, blk, 0, stream>>>(T, A, R1, wq1);
  pack_cw_kernel<F0_ * FNEXT><<<(F0_ * FNEXT / 2) * FNEXT / 256, blk, 0, stream>>>(P, cw1);
  cin_gemm_kernel<FNEXT, 2><<<2 * (NR / 64), blk, 0, stream>>>(R2, A, V, T, P);
  // ---- final sum over embedding dim (+ split-K partials) ----
  reduce_k_kernel<<<(B_ * 2 * FNEXT) / 256, blk, 0, stream>>>((float*)d_out, R1, R2, R2 + nQ);
}